// Net_27762668601788
// MI455X (gfx1250) — compile-verified
//
#include <hip/hip_runtime.h>
#include <math.h>

// Factorized NNConv GNN for MI455X (gfx1250, wave32).
//   Layer l:  T[N, 6*cout] = h[N, cin] @ pack(W2_0..W2_3, B2, root)   (fp32 WMMA GEMM)
//             hnext = T[:,5] + bias ; scatter-add over edges of
//                     sum_k relu(ea*w1_k+b1_k) * T[src, k] + T[src, 4]
//             h = relu(hnext)
// Workspace: hA (N*256) + hB (N*256) + T (N*1536) + Wpack (49152) floats ~ 164 MB.

typedef __attribute__((ext_vector_type(2))) float v2f;
typedef __attribute__((ext_vector_type(8))) float v8f;

static inline int cdiv(int a, int b) { return (a + b - 1) / b; }

// ---------------------------------------------------------------------------
// C[N,M] = A[N,K] @ B[K,M], fp32, via V_WMMA_F32_16X16X4_F32.
// Requires N%16==0, M%16==0, K%4==0. One 16x16 tile per wave.
// fp32 A layout: lane<16 -> row=lane, holds K=k,k+1 ; lane>=16 -> row=lane-16,
// holds K=k+2,k+3. B mirrored with lane = output column. C: 8 VGPRs,
// lane<16 -> col=lane rows r..r ; lane>=16 -> rows r+8.
// ---------------------------------------------------------------------------
__global__ __launch_bounds__(128) void wmma_gemm_f32(
    const float* __restrict__ A, const float* __restrict__ B,
    float* __restrict__ C, int Nn, int K, int M) {
  const int wave = threadIdx.x >> 5;            // 4 waves / block
  const int lane = threadIdx.x & 31;
  const int tile = blockIdx.x * 4 + wave;
  const int tilesM = M >> 4;
  const int row0 = (tile / tilesM) << 4;        // wave-uniform
  const int m0 = (tile % tilesM) << 4;
  if (row0 >= Nn) return;                       // uniform branch: EXEC stays all-1s

  const int l15 = lane & 15;
  const int hi = lane >> 4;                     // 0: K=k,k+1  1: K=k+2,k+3
  const int koff = hi << 1;

  v8f acc = {};
  const float* __restrict__ arow = A + (size_t)(row0 + l15) * K + koff;
  const int colB = m0 + l15;

  for (int k = 0; k < K; k += 4) {
    v2f a = *(const v2f*)(arow + k);            // A[row, k+koff .. k+koff+1]
    v2f b;
    b.x = B[(size_t)(k + koff) * M + colB];     // B[k+koff,   col]
    b.y = B[(size_t)(k + koff + 1) * M + colB]; // B[k+koff+1, col]
    acc = __builtin_amdgcn_wmma_f32_16x16x4_f32(
        /*neg_a=*/false, a, /*neg_b=*/false, b,
        /*c_mod=*/(short)0, acc, /*reuse_a=*/false, /*reuse_b=*/false);
  }

  const int colC = m0 + l15;
  const int rbase = row0 + (hi << 3);
#pragma unroll
  for (int r = 0; r < 8; ++r)
    C[(size_t)(rbase + r) * M + colC] = acc[r];
}

// Build Wpack[cin, 6*cout]: cols [0..4*cout) = W2_k, [4..5) = B2, [5..6) = root.
__global__ void pack_w(const float* __restrict__ w2, const float* __restrict__ b2,
                       const float* __restrict__ root, float* __restrict__ Wp,
                       int cin, int cout) {
  int idx = blockIdx.x * blockDim.x + threadIdx.x;
  int tot = cin * 6 * cout;
  if (idx >= tot) return;
  int i = idx / (6 * cout);
  int rem = idx - i * 6 * cout;
  int k = rem / cout;
  int o = rem - k * cout;
  float v;
  if (k < 4)      v = w2[(size_t)k * cin * cout + (size_t)i * cout + o];
  else if (k == 4) v = b2[(size_t)i * cout + o];
  else             v = root[(size_t)i * cout + o];
  Wp[idx] = v;
}

// hnext[n,o] = T[n, 5*cout + o] + bias[o]   (root term + bias: the accumulator base)
__global__ void fill_base(const float* __restrict__ T, const float* __restrict__ bias,
                          float* __restrict__ hnext, int Nn, int cout, int M) {
  int idx = blockIdx.x * blockDim.x + threadIdx.x;
  if (idx >= Nn * cout) return;
  int n = idx / cout, o = idx - n * cout;
  hnext[(size_t)n * cout + o] = T[(size_t)n * M + 5 * cout + o] + bias[o];
}

// Per (edge, o): msg = sum_k relu(ea*w1k+b1k)*T[src,k*cout+o] + T[src,4*cout+o]
// scatter-add into hnext[dst].
__global__ void edge_msg(const float* __restrict__ T, const float* __restrict__ eattr,
                         const int* __restrict__ src, const int* __restrict__ dst,
                         const float* __restrict__ w1, const float* __restrict__ b1,
                         float* __restrict__ hnext, int Ee, int cout, int M) {
  int idx = blockIdx.x * blockDim.x + threadIdx.x;
  if (idx >= Ee * cout) return;
  int e = idx / cout, o = idx - e * cout;
  float ea = eattr[e];
  float e0 = fmaxf(fmaf(ea, w1[0], b1[0]), 0.f);
  float e1 = fmaxf(fmaf(ea, w1[1], b1[1]), 0.f);
  float e2 = fmaxf(fmaf(ea, w1[2], b1[2]), 0.f);
  float e3 = fmaxf(fmaf(ea, w1[3], b1[3]), 0.f);
  const float* t = T + (size_t)src[e] * M + o;
  float msg = e0 * t[0] + e1 * t[cout] + e2 * t[2 * cout] + e3 * t[3 * cout]
              + t[4 * cout];
  atomicAdd(&hnext[(size_t)dst[e] * cout + o], msg);
}

__global__ void relu_ip(float* __restrict__ h, int n) {
  int i = blockIdx.x * blockDim.x + threadIdx.x;
  if (i < n) h[i] = fmaxf(h[i], 0.f);
}

// For each first-node-of-graph i: out[batch[i]] = sigmoid(h[i+1,:] @ lin_w + lin_b)
__global__ void final_out(const float* __restrict__ h, const int* __restrict__ batch,
                          const float* __restrict__ lin_w, const float* __restrict__ lin_b,
                          float* __restrict__ out, int Nn) {
  int i = blockIdx.x * blockDim.x + threadIdx.x;
  if (i >= Nn) return;
  bool first = (i == 0) || (batch[i] != batch[i - 1]);
  if (!first) return;
  int j = (i + 1 < Nn) ? i + 1 : i;
  float s = lin_b[0];
#pragma unroll
  for (int c = 0; c < 32; ++c) s = fmaf(h[(size_t)j * 32 + c], lin_w[c], s);
  out[batch[i]] = 1.f / (1.f + expf(-s));
}

extern "C" void kernel_launch(void* const* d_in, const int* in_sizes, int n_in,
                              void* d_out, int out_size, void* d_ws, size_t ws_size,
                              hipStream_t stream) {
  const float* x     = (const float*)d_in[0];
  const int*   eidx  = (const int*)d_in[1];
  const float* eattr = (const float*)d_in[2];
  const int*   batch = (const int*)d_in[3];
  const int N = in_sizes[0] / 4;
  const int E = in_sizes[2];
  const int* src = eidx;
  const int* dst = eidx + E;

  float* ws = (float*)d_ws;
  float* hA = ws;
  float* hB = hA + (size_t)N * 256;
  float* T  = hB + (size_t)N * 256;
  float* Wp = T + (size_t)N * 1536;

  const int cins[4]  = {4, 16, 32, 256};
  const int couts[4] = {16, 32, 256, 32};

  const float* hin = x;
  float* hout = hA;
  for (int l = 0; l < 4; ++l) {
    const float* w1   = (const float*)d_in[4 + 6 * l + 0];
    const float* b1   = (const float*)d_in[4 + 6 * l + 1];
    const float* w2   = (const float*)d_in[4 + 6 * l + 2];
    const float* b2   = (const float*)d_in[4 + 6 * l + 3];
    const float* root = (const float*)d_in[4 + 6 * l + 4];
    const float* bias = (const float*)d_in[4 + 6 * l + 5];
    const int cin = cins[l], cout = couts[l], M = 6 * cout;

    pack_w<<<cdiv(cin * M, 256), 256, 0, stream>>>(w2, b2, root, Wp, cin, cout);

    const int tiles = (N / 16) * (M / 16);
    wmma_gemm_f32<<<cdiv(tiles, 4), 128, 0, stream>>>(hin, Wp, T, N, cin, M);

    fill_base<<<cdiv(N * cout, 256), 256, 0, stream>>>(T, bias, hout, N, cout, M);

    edge_msg<<<cdiv(E * cout, 256), 256, 0, stream>>>(T, eattr, src, dst, w1, b1,
                                                      hout, E, cout, M);

    relu_ip<<<cdiv(N * cout, 256), 256, 0, stream>>>(hout, N * cout);

    hin = hout;
    hout = (hout == hA) ? hB : hA;
  }

  const float* lin_w = (const float*)d_in[28];
  const float* lin_b = (const float*)d_in[29];
  final_out<<<cdiv(N, 256), 256, 0, stream>>>(hin, batch, lin_w, lin_b,
                                              (float*)d_out, N);
}